// SimpleSelectiveScan_20993800143235
// MI455X (gfx1250) — compile-verified
//
#include <hip/hip_runtime.h>
#include <cstdint>
#include <cstddef>

// Problem constants (reference: B=4, L=4096, D=1024, fp32)
#define BATCH 4
#define SEQ   4096
#define DIM   1024
#define NCH   (BATCH * DIM)     // 4096 independent scan channels
#define NC    64                // chunks along L
#define TC    (SEQ / NC)        // 64 timesteps per chunk
#define TT    16                // timesteps per LDS tile
#define NT    (TC / TT)         // 4 tiles per chunk
#define CHB   128               // channels per block (= blockDim.x, 4 waves)
#define NBLK_CH (NCH / CHB)     // 32

// ---------------------------------------------------------------------------
// CDNA5 async global->LDS copy (ASYNCcnt-tracked), per-lane 16B.
// Builtin takes v4i pointers: (AS1 src, AS3 dst, imm offset, imm cpol).
// LDS dest address: low 32 bits of the generic pointer == LDS byte offset.
// ---------------------------------------------------------------------------
typedef int v4i_vs __attribute__((vector_size(16)));
typedef __attribute__((address_space(1))) v4i_vs* v4i_g_ptr;
typedef __attribute__((address_space(3))) v4i_vs* v4i_l_ptr;

__device__ __forceinline__ uint32_t lds_off_u32(const void* p) {
    return (uint32_t)(uintptr_t)p;
}

__device__ __forceinline__ void async_load_b128(uint32_t lds_off, const float* gaddr) {
#if __has_builtin(__builtin_amdgcn_global_load_async_to_lds_b128)
    __builtin_amdgcn_global_load_async_to_lds_b128(
        (v4i_g_ptr)(uintptr_t)gaddr,
        (v4i_l_ptr)(uintptr_t)lds_off,
        0, 0);
#else
    asm volatile("global_load_async_to_lds_b128 %0, %1, off"
                 :: "v"(lds_off), "v"(gaddr)
                 : "memory");
#endif
}

#if __has_builtin(__builtin_amdgcn_s_wait_asynccnt)
#define WAIT_ASYNC(N) __builtin_amdgcn_s_wait_asynccnt(N)
#else
#define WAIT_ASYNC(N) asm volatile("s_wait_asynccnt %0" :: "n"(N) : "memory")
#endif

__device__ __forceinline__ float sigmoidf_dev(float v) {
    return 1.0f / (1.0f + __expf(-v));
}

// ---------------------------------------------------------------------------
// Phase 1: per-chunk aggregates (A = prod a_t, U = chunk-local state, init 0)
// grid = (NBLK_CH, NC), block = CHB. Double-buffered async LDS staging of
// delta / b_term / x tiles (TT x CHB each). 12 async instrs per wave per tile.
// ---------------------------------------------------------------------------
__global__ __launch_bounds__(CHB)
void scan_phase1(const float* __restrict__ delta,
                 const float* __restrict__ b_term,
                 const float* __restrict__ x,
                 float* __restrict__ Aagg,
                 float* __restrict__ Uagg)
{
    __shared__ __attribute__((aligned(16))) float sD[2][TT * CHB];
    __shared__ __attribute__((aligned(16))) float sB[2][TT * CHB];
    __shared__ __attribute__((aligned(16))) float sX[2][TT * CHB];

    const int tid   = threadIdx.x;
    const int chunk = blockIdx.y;
    const int ch0   = blockIdx.x * CHB;
    const int b     = ch0 / DIM;            // CHB divides DIM -> single batch
    const int d0    = ch0 % DIM;
    const int t0    = chunk * TC;

    // element index of (t0, d0+chL): ((b*SEQ + t0)*DIM + d0)
    const size_t base = ((size_t)b * SEQ + t0) * (size_t)DIM + d0;

    const int tt_l = tid >> 5;              // 0..3
    const int ch_l = (tid & 31) * 4;        // 0..124, 16B per lane

    const uint32_t ldsD0 = lds_off_u32(&sD[0][0]);
    const uint32_t ldsB0 = lds_off_u32(&sB[0][0]);
    const uint32_t ldsX0 = lds_off_u32(&sX[0][0]);
    const uint32_t bufStride = (uint32_t)(TT * CHB * sizeof(float));

    auto issue_tile = [&](int tile, int buf) {
        const size_t tb = base + (size_t)(tile * TT) * DIM;
        const uint32_t lb = (uint32_t)buf * bufStride;
#pragma unroll
        for (int j = 0; j < 4; ++j) {
            const int tt = j * 4 + tt_l;
            const size_t  g = tb + (size_t)tt * DIM + ch_l;
            const uint32_t l = lb + (uint32_t)((tt * CHB + ch_l) * sizeof(float));
            async_load_b128(ldsD0 + l, delta  + g);
            async_load_b128(ldsB0 + l, b_term + g);
            async_load_b128(ldsX0 + l, x      + g);
        }
    };

    issue_tile(0, 0);
    issue_tile(1, 1);

    float A = 1.0f, U = 0.0f;
#pragma unroll
    for (int tile = 0; tile < NT; ++tile) {
        const int buf = tile & 1;
        if (tile < NT - 1) { WAIT_ASYNC(12); } else { WAIT_ASYNC(0); }
        __syncthreads();                    // this tile visible to all waves
#pragma unroll
        for (int tt = 0; tt < TT; ++tt) {
            const int idx = tt * CHB + tid;
            const float a = sigmoidf_dev(sD[buf][idx]);
            const float u = tanhf(sB[buf][idx]) * sX[buf][idx];
            A *= a;
            U = fmaf(a, U, u);
        }
        __syncthreads();                    // done reading buf before overwrite
        if (tile + 2 < NT) issue_tile(tile + 2, buf);
    }

    const int ch = ch0 + tid;
    Aagg[chunk * NCH + ch] = A;
    Uagg[chunk * NCH + ch] = U;
}

// ---------------------------------------------------------------------------
// Phase 2: per-channel exclusive scan over chunk aggregates.
// Init[c] = state entering chunk c;  s' = A_c * s + U_c.
// 4096 threads total; tiny (3 MB traffic).
// ---------------------------------------------------------------------------
__global__ __launch_bounds__(256)
void scan_phase2(const float* __restrict__ Aagg,
                 const float* __restrict__ Uagg,
                 float* __restrict__ Init)
{
    const int ch = blockIdx.x * 256 + threadIdx.x;
    float s = 0.0f;
#pragma unroll 8
    for (int c = 0; c < NC; ++c) {
        Init[c * NCH + ch] = s;
        s = fmaf(Aagg[c * NCH + ch], s, Uagg[c * NCH + ch]);
    }
}

// ---------------------------------------------------------------------------
// Phase 3: recompute scan within each chunk from its carried-in state,
// produce out = tanh(c)*state + skip*x. Stages 4 tensors per tile
// (16 async instrs per wave per tile). Output stores are non-temporal so the
// 67 MB write-once stream does not evict the phase1->phase3 L2 reuse set
// (delta/b/x: 201 MB vs 192 MB L2).
// ---------------------------------------------------------------------------
__global__ __launch_bounds__(CHB)
void scan_phase3(const float* __restrict__ delta,
                 const float* __restrict__ b_term,
                 const float* __restrict__ c_term,
                 const float* __restrict__ x,
                 const float* __restrict__ skip,
                 const float* __restrict__ Init,
                 float* __restrict__ out)
{
    __shared__ __attribute__((aligned(16))) float sD[2][TT * CHB];
    __shared__ __attribute__((aligned(16))) float sB[2][TT * CHB];
    __shared__ __attribute__((aligned(16))) float sC[2][TT * CHB];
    __shared__ __attribute__((aligned(16))) float sX[2][TT * CHB];

    const int tid   = threadIdx.x;
    const int chunk = blockIdx.y;
    const int ch0   = blockIdx.x * CHB;
    const int b     = ch0 / DIM;
    const int d0    = ch0 % DIM;
    const int t0    = chunk * TC;

    const size_t base = ((size_t)b * SEQ + t0) * (size_t)DIM + d0;

    const int tt_l = tid >> 5;
    const int ch_l = (tid & 31) * 4;

    const uint32_t ldsD0 = lds_off_u32(&sD[0][0]);
    const uint32_t ldsB0 = lds_off_u32(&sB[0][0]);
    const uint32_t ldsC0 = lds_off_u32(&sC[0][0]);
    const uint32_t ldsX0 = lds_off_u32(&sX[0][0]);
    const uint32_t bufStride = (uint32_t)(TT * CHB * sizeof(float));

    auto issue_tile = [&](int tile, int buf) {
        const size_t tb = base + (size_t)(tile * TT) * DIM;
        const uint32_t lb = (uint32_t)buf * bufStride;
#pragma unroll
        for (int j = 0; j < 4; ++j) {
            const int tt = j * 4 + tt_l;
            const size_t  g = tb + (size_t)tt * DIM + ch_l;
            const uint32_t l = lb + (uint32_t)((tt * CHB + ch_l) * sizeof(float));
            async_load_b128(ldsD0 + l, delta  + g);
            async_load_b128(ldsB0 + l, b_term + g);
            async_load_b128(ldsC0 + l, c_term + g);
            async_load_b128(ldsX0 + l, x      + g);
        }
    };

    issue_tile(0, 0);
    issue_tile(1, 1);

    const int ch = ch0 + tid;
    float s        = Init[chunk * NCH + ch];
    const float sk = skip[d0 + tid];

#pragma unroll
    for (int tile = 0; tile < NT; ++tile) {
        const int buf = tile & 1;
        if (tile < NT - 1) { WAIT_ASYNC(16); } else { WAIT_ASYNC(0); }
        __syncthreads();
#pragma unroll
        for (int tt = 0; tt < TT; ++tt) {
            const int idx = tt * CHB + tid;
            const float xv = sX[buf][idx];
            const float a  = sigmoidf_dev(sD[buf][idx]);
            const float u  = tanhf(sB[buf][idx]) * xv;
            s = fmaf(a, s, u);
            const float o = fmaf(tanhf(sC[buf][idx]), s, sk * xv);
            __builtin_nontemporal_store(o, &out[base + (size_t)(tile * TT + tt) * DIM + tid]);
        }
        __syncthreads();
        if (tile + 2 < NT) issue_tile(tile + 2, buf);
    }
}

// ---------------------------------------------------------------------------
// Launch: inputs in dict order: x, delta, b_term, c_term, skip. Output fp32.
// Workspace: Aagg | Uagg | Init, each NC*NCH floats (3 MB total).
// ---------------------------------------------------------------------------
extern "C" void kernel_launch(void* const* d_in, const int* in_sizes, int n_in,
                              void* d_out, int out_size, void* d_ws, size_t ws_size,
                              hipStream_t stream) {
    const float* x      = (const float*)d_in[0];
    const float* delta  = (const float*)d_in[1];
    const float* b_term = (const float*)d_in[2];
    const float* c_term = (const float*)d_in[3];
    const float* skip   = (const float*)d_in[4];
    float* out = (float*)d_out;

    float* Aagg = (float*)d_ws;
    float* Uagg = Aagg + (size_t)NC * NCH;
    float* Init = Uagg + (size_t)NC * NCH;

    dim3 grid13(NBLK_CH, NC);
    scan_phase1<<<grid13, CHB, 0, stream>>>(delta, b_term, x, Aagg, Uagg);
    scan_phase2<<<NCH / 256, 256, 0, stream>>>(Aagg, Uagg, Init);
    scan_phase3<<<grid13, CHB, 0, stream>>>(delta, b_term, c_term, x, skip, Init, out);
}